// lowrank_expirimental_RNN_77266461655424
// MI455X (gfx1250) — compile-verified
//
#include <hip/hip_runtime.h>
#include <hip/hip_bf16.h>

typedef __attribute__((ext_vector_type(2))) float v2f;
typedef __attribute__((ext_vector_type(8))) float v8f;
typedef __attribute__((ext_vector_type(4))) int v4i;

#define B_DIM 64
#define T_DIM 512
#define I_DIM 128
#define H_DIM 1024
#define O_DIM 64
#define R_DIM 16
#define DT_C  0.1f

static __device__ __forceinline__ v8f wmma_f32(v2f a, v2f b, v8f c) {
  // V_WMMA_F32_16X16X4_F32: D = A(16x4,f32) * B(4x16,f32) + C(16x16,f32)
  return __builtin_amdgcn_wmma_f32_16x16x4_f32(false, a, false, b, (short)0, c,
                                               false, false);
}

// ---- async global->LDS copy (ASYNCcnt path), with synchronous fallback ----
#if defined(__has_builtin) && __has_builtin(__builtin_amdgcn_global_load_async_to_lds_b128)
#define HAVE_ASYNC_LDS 1
typedef __attribute__((address_space(1))) v4i* as1_v4i_p;  // global v4i*
typedef __attribute__((address_space(3))) v4i* as3_v4i_p;  // LDS v4i*
static __device__ __forceinline__ void async_cp16(const float* g, float* l) {
  // integer round-trip: generic->global is bit-identical; generic->LDS is the
  // low 32 bits; p3 is 32-bit.
  __builtin_amdgcn_global_load_async_to_lds_b128(
      (as1_v4i_p)(unsigned long long)g,
      (as3_v4i_p)(unsigned int)(unsigned long long)l, 0, 0);
}
#if __has_builtin(__builtin_amdgcn_s_wait_asynccnt)
static __device__ __forceinline__ void wait_async0() {
  __builtin_amdgcn_s_wait_asynccnt(0);
}
#else
static __device__ __forceinline__ void wait_async0() {
  asm volatile("s_wait_asynccnt 0x0" ::: "memory");
}
#endif
#else
#define HAVE_ASYNC_LDS 0
static __device__ __forceinline__ void async_cp16(const float* g, float* l) {
  *(float4*)l = *(const float4*)g;  // synchronous fallback
}
static __device__ __forceinline__ void wait_async0() {}
#endif

// ---------------------------------------------------------------------------
// Kernel 1: x_proj[b,t,h] = sum_i inputs[b,t,i] * W_ih[h,i]
// GEMM: M = B*T = 32768, N = H = 1024, K = I = 128 (full K staged in LDS).
// Block = 256 threads (8 waves), block tile 32(M) x 64(N), wave tile 16x16.
// A tile streamed in via GLOBAL_LOAD_ASYNC_TO_LDS_B128.
// ---------------------------------------------------------------------------
__global__ __launch_bounds__(256) void xproj_gemm(const float* __restrict__ inp,
                                                  const float* __restrict__ W_ih,
                                                  float* __restrict__ xp) {
  __shared__ float As[32][I_DIM];      // 16 KB : inputs tile, row-major in K
  __shared__ float Bs[I_DIM][64 + 4];  // 34 KB : W_ih^T tile (transposed load)

  const int tid  = threadIdx.x;
  const int wv   = tid >> 5;
  const int lane = tid & 31;
  const int m0 = blockIdx.x * 32;
  const int n0 = blockIdx.y * 64;

  // A tile: 4096 floats = 1024 x b128, async into LDS
  for (int idx = tid; idx < 32 * (I_DIM / 4); idx += 256) {
    int r = idx >> 5, c4 = (idx & 31) * 4;
    async_cp16(&inp[(size_t)(m0 + r) * I_DIM + c4], &As[r][c4]);
  }
  // B tile: transpose on load (element reshuffle -> synchronous)
  for (int idx = tid; idx < 64 * I_DIM; idx += 256) {
    int nn = idx >> 7, k = idx & (I_DIM - 1);
    Bs[k][nn] = W_ih[(size_t)(n0 + nn) * I_DIM + k];  // Bs[k][n]
  }
  wait_async0();
  __syncthreads();

  const int mt = wv >> 2;              // 0..1
  const int nt = wv & 3;               // 0..3
  const int rc = lane & 15;            // row (A) / col (B,D)
  const int kb = (lane >> 4) << 1;     // K sub-offset {0,2} per lane half
  const int rbase = (lane >> 4) << 3;  // C/D: M offset {0,8} per lane half

  v8f acc = {};
  for (int k = 0; k < I_DIM; k += 4) {
    v2f a, b;
    a.x = As[mt * 16 + rc][k + kb];
    a.y = As[mt * 16 + rc][k + kb + 1];
    b.x = Bs[k + kb][nt * 16 + rc];
    b.y = Bs[k + kb + 1][nt * 16 + rc];
    acc = wmma_f32(a, b, acc);
  }

#pragma unroll
  for (int i = 0; i < 8; ++i) {
    int row = m0 + mt * 16 + rbase + i;
    int col = n0 + nt * 16 + rc;
    xp[(size_t)row * H_DIM + col] = acc[i];
  }
}

// ---------------------------------------------------------------------------
// Kernel 2: persistent sequential scan. 4 blocks; block b owns batch rows
// [16b, 16b+16) for all 512 steps. Hidden state, n, inv_tau and a double-
// buffered x_proj tile live in LDS (~265 KB of the WGP's 320 KB). Per step:
//   p   = tanh(h) @ n^T            (16x16, K=1024 split over 8 waves, ds_add)
//   q   = (p/H) @ m^T              (16x1024, fused leaky-Euler h update)
//   out = tanh(h_new) @ W_ho^T + b (16x64,  K split over 8 waves, ds_add)
// x_proj tile for step t+1 streams in via async-to-LDS while t computes.
// ---------------------------------------------------------------------------
__global__ __launch_bounds__(256) void lowrank_rnn_scan(
    const float* __restrict__ xp,    // (B*T, H) precomputed projections
    const float* __restrict__ h0,    // (B, H)
    const float* __restrict__ m,     // (H, R)
    const float* __restrict__ n,     // (R, H)
    const float* __restrict__ W_ho,  // (O, H)
    const float* __restrict__ b_ho,  // (O,)
    const float* __restrict__ taus,  // (H,)
    float* __restrict__ hT,          // (B, H)
    float* __restrict__ outs) {      // (B, T, O)
  __shared__ float h_s[16][H_DIM];      //  64 KB  hidden state
  __shared__ float n_s[R_DIM][H_DIM];   //  64 KB  n (natural layout)
  __shared__ float xb_s[2][16][H_DIM];  // 128 KB  double-buffered x_proj tile
  __shared__ float invtau_s[H_DIM];     //   4 KB
  __shared__ float p_s[16][16];         //   1 KB  low-rank bottleneck
  __shared__ float outacc[16][O_DIM];   //   4 KB  per-step output accum
  __shared__ float bho_s[O_DIM];

  const int tid   = threadIdx.x;
  const int wv    = tid >> 5;  // 0..7
  const int lane  = tid & 31;
  const int rc    = lane & 15;
  const int kb    = (lane >> 4) << 1;
  const int rbase = (lane >> 4) << 3;
  const int b0    = blockIdx.x * 16;

  for (int idx = tid; idx < 16 * H_DIM; idx += 256) {
    int r = idx >> 10, c = idx & (H_DIM - 1);
    h_s[r][c] = h0[(size_t)(b0 + r) * H_DIM + c];
    n_s[r][c] = n[idx];  // (R=16, H) matches 16 x H_DIM footprint
  }
  for (int idx = tid; idx < H_DIM; idx += 256) invtau_s[idx] = 1.0f / taus[idx];
  if (tid < O_DIM) bho_s[tid] = b_ho[tid];

  // kick off async staging of the t=0 x_proj tile (16 x b128 per thread)
  for (int idx = tid; idx < 16 * (H_DIM / 4); idx += 256) {
    int r = idx >> 8, c4 = (idx & 255) * 4;
    async_cp16(&xp[((size_t)(b0 + r) * T_DIM + 0) * H_DIM + c4],
               &xb_s[0][r][c4]);
  }
  __syncthreads();

  const float invH = 1.0f / (float)H_DIM;

  for (int t = 0; t < T_DIM; ++t) {
    const int cur = t & 1;
    // zero the cross-wave accumulators
    p_s[tid >> 4][tid & 15] = 0.0f;
    for (int idx = tid; idx < 16 * O_DIM; idx += 256)
      outacc[idx >> 6][idx & 63] = 0.0f;
    __syncthreads();

    // ---- p = tanh(h) @ n^T : each wave reduces a 128-wide K slice ----
    {
      v8f cp = {};
      const int kbase = wv * 128;
      for (int kk = 0; kk < 128; kk += 4) {
        int k = kbase + kk;
        v2f a, b;
        a.x = tanhf(h_s[rc][k + kb]);
        a.y = tanhf(h_s[rc][k + kb + 1]);
        b.x = n_s[rc][k + kb];  // B[k][j] = n[j][k]
        b.y = n_s[rc][k + kb + 1];
        cp = wmma_f32(a, b, cp);
      }
#pragma unroll
      for (int i = 0; i < 8; ++i) atomicAdd(&p_s[rbase + i][rc], cp[i]);
    }
    wait_async0();   // x_proj tile t fully resident in xb_s[cur]
    __syncthreads();

    // ---- q = (p/H) @ m^T, fused with h_new = h + dt*(-h/tau + q + xp) ----
    {
      v2f pa[4];
#pragma unroll
      for (int ki = 0; ki < 4; ++ki) {
        pa[ki].x = p_s[rc][ki * 4 + kb] * invH;
        pa[ki].y = p_s[rc][ki * 4 + kb + 1] * invH;
      }
      for (int ct = wv; ct < H_DIM / 16; ct += 8) {  // wave-private col tiles
        v8f cq = {};
#pragma unroll
        for (int ki = 0; ki < 4; ++ki) {
          v2f b;
          b.x = m[(size_t)(ct * 16 + rc) * R_DIM + ki * 4 + kb];  // m[h][k]
          b.y = m[(size_t)(ct * 16 + rc) * R_DIM + ki * 4 + kb + 1];
          cq = wmma_f32(pa[ki], b, cq);
        }
#pragma unroll
        for (int i = 0; i < 8; ++i) {
          int r = rbase + i;
          int c = ct * 16 + rc;
          float hv = h_s[r][c];
          float xv = xb_s[cur][r][c];
          float hn = hv + DT_C * (-hv * invtau_s[c] + cq[i] + xv);
          h_s[r][c] = hn;
        }
      }
    }
    // stream next timestep's x_proj tile into the other buffer (ASYNCcnt)
    if (t + 1 < T_DIM) {
      for (int idx = tid; idx < 16 * (H_DIM / 4); idx += 256) {
        int r = idx >> 8, c4 = (idx & 255) * 4;
        async_cp16(&xp[((size_t)(b0 + r) * T_DIM + (t + 1)) * H_DIM + c4],
                   &xb_s[cur ^ 1][r][c4]);
      }
    }
    __syncthreads();

    // ---- out = tanh(h_new) @ W_ho^T : K split over waves, 4 col tiles ----
    {
      v8f co[4] = {};
      const int kbase = wv * 128;
      for (int kk = 0; kk < 128; kk += 4) {
        int k = kbase + kk;
        v2f a;
        a.x = tanhf(h_s[rc][k + kb]);
        a.y = tanhf(h_s[rc][k + kb + 1]);
#pragma unroll
        for (int to = 0; to < 4; ++to) {
          v2f b;
          b.x = W_ho[(size_t)(to * 16 + rc) * H_DIM + k + kb];
          b.y = W_ho[(size_t)(to * 16 + rc) * H_DIM + k + kb + 1];
          co[to] = wmma_f32(a, b, co[to]);
        }
      }
#pragma unroll
      for (int to = 0; to < 4; ++to)
#pragma unroll
        for (int i = 0; i < 8; ++i)
          atomicAdd(&outacc[rbase + i][to * 16 + rc], co[to][i]);
    }
    __syncthreads();

    // ---- emit outputs for step t ----
    for (int idx = tid; idx < 16 * O_DIM; idx += 256) {
      int r = idx >> 6, o = idx & 63;
      outs[((size_t)(b0 + r) * T_DIM + t) * O_DIM + o] = outacc[r][o] + bho_s[o];
    }
    __syncthreads();
  }

  for (int idx = tid; idx < 16 * H_DIM; idx += 256) {
    int r = idx >> 10, c = idx & (H_DIM - 1);
    hT[(size_t)(b0 + r) * H_DIM + c] = h_s[r][c];
  }
}

extern "C" void kernel_launch(void* const* d_in, const int* in_sizes, int n_in,
                              void* d_out, int out_size, void* d_ws,
                              size_t ws_size, hipStream_t stream) {
  const float* inputs = (const float*)d_in[0];
  const float* hidden = (const float*)d_in[1];
  const float* m      = (const float*)d_in[2];
  const float* n      = (const float*)d_in[3];
  const float* W_ih   = (const float*)d_in[4];
  const float* W_ho   = (const float*)d_in[5];
  const float* b_ho   = (const float*)d_in[6];
  const float* taus   = (const float*)d_in[7];

  // d_out = [ h_T (B,H) | outs (B,T,O) ] in reference return order
  float* hT   = (float*)d_out;
  float* outs = (float*)d_out + (size_t)B_DIM * H_DIM;
  float* xp   = (float*)d_ws;  // (B*T, H) = 128 MB scratch

  dim3 g1((B_DIM * T_DIM) / 32, H_DIM / 64);
  xproj_gemm<<<g1, 256, 0, stream>>>(inputs, W_ih, xp);
  lowrank_rnn_scan<<<B_DIM / 16, 256, 0, stream>>>(xp, hidden, m, n, W_ho, b_ho,
                                                   taus, hT, outs);
  (void)in_sizes; (void)n_in; (void)out_size; (void)ws_size;
}